// ResidualVQVAE_2662879724286
// MI455X (gfx1250) — compile-verified
//
#include <hip/hip_runtime.h>

// ---------------------------------------------------------------------------
// Residual VQ-VAE forward for MI455X (gfx1250, wave32, WMMA).
// - All GEMMs use V_WMMA_F32_16X16X4_F32 (full f32 precision; argmin over
//   near-tied codebook distances is not bf16/f16 safe).
// - The dominant VQ distance GEMM (8 x [8192x8192x128] = 137 GFLOP) stages
//   codebook chunks into LDS with GLOBAL_LOAD_ASYNC_TO_LDS_B128 (ASYNCcnt),
//   double-buffered, shared by all 4 waves of the workgroup.
// ---------------------------------------------------------------------------

typedef __attribute__((ext_vector_type(2))) float v2f;
typedef __attribute__((ext_vector_type(8))) float v8f;
typedef __attribute__((ext_vector_type(4))) int   v4i;

#define NROWS  8192
#define INDIM  256
#define CDIM   128
#define KCODES 8192
#define MBOOK  8
#define SROW   132   // padded LDS row stride (floats): 132*n mod 64 = 4n -> no bank conflicts

#if __has_builtin(__builtin_amdgcn_global_load_async_to_lds_b128)
#define HAVE_ASYNC_LDS 1
#else
#define HAVE_ASYNC_LDS 0
#endif

#if HAVE_ASYNC_LDS
typedef __attribute__((address_space(1))) v4i* gv4i_ptr;   // global v4i*
typedef __attribute__((address_space(3))) v4i* lv4i_ptr;   // LDS v4i*

__device__ __forceinline__ void async_ld16(const float* g, float* s)
{
    // 16 bytes per lane, global -> LDS, tracked by ASYNCcnt
    __builtin_amdgcn_global_load_async_to_lds_b128(
        (gv4i_ptr)(v4i*)g, (lv4i_ptr)(v4i*)s, 0, 0);
}
#endif

__device__ __forceinline__ void wait_async0()
{
#if __has_builtin(__builtin_amdgcn_s_wait_asynccnt)
    __builtin_amdgcn_s_wait_asynccnt(0);
#else
    asm volatile("s_wait_asynccnt 0x0" ::: "memory");
#endif
}

// ---------------------------------------------------------------------------
// Fused dense layer: Y = act(bn(X @ W + b))   (bn/relu optional)
// One wave per 16x16 output tile; K consumed 4 at a time by WMMA f32 16x16x4.
// A-frag: lane l<16 holds row l, K=c+0/c+1; lanes 16-31 row l-16, K=c+2/c+3.
// C/D: VGPR i holds row (i + 8*(lane>>4)), col = lane&15.
// ---------------------------------------------------------------------------
__global__ __launch_bounds__(128) void dense_wmma(
    const float* __restrict__ X, const float* __restrict__ W,
    const float* __restrict__ bias,
    const float* __restrict__ g, const float* __restrict__ be,
    const float* __restrict__ rm, const float* __restrict__ rv,
    float* __restrict__ Y, int C, int O, int do_relu)
{
    const int lane  = threadIdx.x & 31;
    const int wid   = threadIdx.x >> 5;
    const int tile  = blockIdx.x * 4 + wid;
    const int tilesN = O >> 4;
    const int tm    = tile / tilesN;
    const int tn    = tile - tm * tilesN;
    const int l15   = lane & 15;
    const int half  = lane >> 4;
    const int choff = half * 2;
    const int col   = (tn << 4) + l15;

    const float* xrow = X + (size_t)(tm * 16 + l15) * C;
    const float* wcol = W + col;

    v8f acc = {};
    #pragma unroll 8
    for (int c = 0; c < C; c += 4) {
        v2f a, b;
        a.x = xrow[c + choff];
        a.y = xrow[c + choff + 1];
        b.x = wcol[(size_t)(c + choff) * O];
        b.y = wcol[(size_t)(c + choff + 1) * O];
        acc = __builtin_amdgcn_wmma_f32_16x16x4_f32(
            false, a, false, b, (short)0, acc, false, false);
    }

    // fold bias into BN:  y = v*scale + shift
    float scale = 1.0f, shift = bias[col];
    if (g != nullptr) {
        const float inv = rsqrtf(rv[col] + 1e-5f);
        scale = g[col] * inv;
        shift = (bias[col] - rm[col]) * scale + be[col];
    }
    #pragma unroll
    for (int i = 0; i < 8; ++i) {
        float v = acc[i] * scale + shift;
        if (do_relu) v = fmaxf(v, 0.0f);
        Y[(size_t)(tm * 16 + i + half * 8) * O + col] = v;
    }
}

// ---------------------------------------------------------------------------
// ||E_k||^2 per codebook row.
// ---------------------------------------------------------------------------
__global__ __launch_bounds__(256) void ee_norm(
    const float* __restrict__ E, float* __restrict__ ee)
{
    const int k = blockIdx.x * 256 + threadIdx.x;
    const float4* p = (const float4*)(E + (size_t)k * CDIM);
    float s = 0.0f;
    #pragma unroll
    for (int i = 0; i < CDIM / 4; ++i) {
        const float4 v = p[i];
        s += v.x * v.x + v.y * v.y + v.z * v.z + v.w * v.w;
    }
    ee[k] = s;
}

// ---------------------------------------------------------------------------
// VQ argmin: score(k) = ||e_k||^2 - 2 * res.e_k   (||res||^2 is row-constant).
// One wave per 16-row residual tile; A fragment (16x128 f32) lives entirely in
// registers across all 512 codebook chunks. Codebook chunks (16x128 = 8KB) are
// async-staged into double-buffered LDS, shared by the 4 waves; 32 WMMAs/chunk.
// Cross-lane argmin via wave32 shfl_xor, tie-break to lower k (jnp.argmin).
// ---------------------------------------------------------------------------
__global__ __launch_bounds__(128) void vq_argmin(
    const float* __restrict__ R, const float* __restrict__ E,
    const float* __restrict__ ee, int* __restrict__ idx_out)
{
    const int tid   = threadIdx.x;
    const int lane  = tid & 31;
    const int wid   = tid >> 5;
    const int tile  = blockIdx.x * 4 + wid;
    const int l15   = lane & 15;
    const int half  = lane >> 4;
    const int choff = half * 2;

    const float* rrow = R + (size_t)(tile * 16 + l15) * CDIM;
    v2f a[32];
    #pragma unroll
    for (int c = 0; c < 32; ++c) {
        a[c].x = rrow[4 * c + choff];
        a[c].y = rrow[4 * c + choff + 1];
    }

    float bestd[8];
    int   bestk[8];
    #pragma unroll
    for (int i = 0; i < 8; ++i) { bestd[i] = 3.402823466e38f; bestk[i] = 0; }

#if HAVE_ASYNC_LDS
    __shared__ float sE[2][16 * SROW];   // double-buffered chunk, padded rows

    // stage chunk 0 (each thread issues 4 x b128 async copies)
    #pragma unroll
    for (int j = 0; j < 4; ++j) {
        const int q   = tid + j * 128;           // quad index 0..511
        const int row = q >> 5;                  // 32 quads per 128-ch row
        const int qc  = q & 31;
        async_ld16(E + (size_t)row * CDIM + qc * 4, &sE[0][row * SROW + qc * 4]);
    }
    wait_async0();
    __syncthreads();

    for (int k0 = 0; k0 < KCODES; k0 += 16) {
        const int buf = (k0 >> 4) & 1;

        if (k0 + 16 < KCODES) {                  // async-prefetch next chunk
            #pragma unroll
            for (int j = 0; j < 4; ++j) {
                const int q   = tid + j * 128;
                const int row = q >> 5;
                const int qc  = q & 31;
                async_ld16(E + (size_t)(k0 + 16 + row) * CDIM + qc * 4,
                           &sE[buf ^ 1][row * SROW + qc * 4]);
            }
        }

        const int k = k0 + l15;
        const float eek = ee[k];
        const float* brow = &sE[buf][l15 * SROW];

        v8f acc = {};
        #pragma unroll
        for (int c = 0; c < 32; ++c) {
            v2f b;
            b.x = brow[4 * c + choff];            // ds_load_b64, conflict-free
            b.y = brow[4 * c + choff + 1];
            acc = __builtin_amdgcn_wmma_f32_16x16x4_f32(
                false, a[c], false, b, (short)0, acc, false, false);
        }
        #pragma unroll
        for (int i = 0; i < 8; ++i) {
            const float d = eek - 2.0f * acc[i];
            if (d < bestd[i]) { bestd[i] = d; bestk[i] = k; }
        }

        wait_async0();                           // next chunk resident in LDS
        __syncthreads();                         // all waves' copies complete
    }
#else
    // fallback: direct global B-fragment loads (WGP$-served, 4x redundant)
    for (int k0 = 0; k0 < KCODES; k0 += 16) {
        const int k = k0 + l15;
        const float* erow = E + (size_t)k * CDIM;
        if (k0 + 16 < KCODES)
            __builtin_prefetch(erow + 16 * CDIM, 0, 1);
        const float eek = ee[k];

        v8f acc = {};
        #pragma unroll
        for (int c = 0; c < 32; ++c) {
            v2f b;
            b.x = erow[4 * c + choff];
            b.y = erow[4 * c + choff + 1];
            acc = __builtin_amdgcn_wmma_f32_16x16x4_f32(
                false, a[c], false, b, (short)0, acc, false, false);
        }
        #pragma unroll
        for (int i = 0; i < 8; ++i) {
            const float d = eek - 2.0f * acc[i];
            if (d < bestd[i]) { bestd[i] = d; bestk[i] = k; }
        }
    }
#endif

    // reduce over the 16 columns held by lanes of the same half (xor 8,4,2,1)
    for (int off = 8; off; off >>= 1) {
        #pragma unroll
        for (int i = 0; i < 8; ++i) {
            const float od = __shfl_xor(bestd[i], off, 32);
            const int   ok = __shfl_xor(bestk[i], off, 32);
            if (od < bestd[i] || (od == bestd[i] && ok < bestk[i])) {
                bestd[i] = od; bestk[i] = ok;
            }
        }
    }
    if (l15 == 0) {
        #pragma unroll
        for (int i = 0; i < 8; ++i)
            idx_out[tile * 16 + i + half * 8] = bestk[i];
    }
}

// ---------------------------------------------------------------------------
// VQ stage finish: res_s[m] = res_in ; ce_s[m] = E[idx] ; res_next = res - ce.
// ---------------------------------------------------------------------------
__global__ __launch_bounds__(256) void vq_finish(
    const float* __restrict__ res_in, const float* __restrict__ E,
    const int* __restrict__ idx,
    float* __restrict__ res_s, float* __restrict__ ce_s,
    float* __restrict__ res_next)
{
    const int t   = blockIdx.x * 256 + threadIdx.x;
    const int row = t >> 5;
    const int c   = (t & 31) << 2;
    const size_t o = (size_t)row * CDIM + c;
    const float4 r = *(const float4*)(res_in + o);
    const int k = idx[row];
    const float4 e = *(const float4*)(E + (size_t)k * CDIM + c);
    *(float4*)(res_s + o) = r;
    *(float4*)(ce_s + o)  = e;
    float4 n;
    n.x = r.x - e.x; n.y = r.y - e.y; n.z = r.z - e.z; n.w = r.w - e.w;
    *(float4*)(res_next + o) = n;
}

// di = ze + sg(zq - ze) == zq == ze - res_final
__global__ __launch_bounds__(256) void di_sub(
    const float* __restrict__ ze, const float* __restrict__ rf,
    float* __restrict__ di)
{
    const int t = blockIdx.x * 256 + threadIdx.x;
    const size_t o = (size_t)t * 4;
    const float4 z = *(const float4*)(ze + o);
    const float4 r = *(const float4*)(rf + o);
    float4 d;
    d.x = z.x - r.x; d.y = z.y - r.y; d.z = z.z - r.z; d.w = z.w - r.w;
    *(float4*)(di + o) = d;
}

// ---------------------------------------------------------------------------
extern "C" void kernel_launch(void* const* d_in, const int* in_sizes, int n_in,
                              void* d_out, int out_size, void* d_ws, size_t ws_size,
                              hipStream_t stream)
{
    const float* x   = (const float*)d_in[0];
    const float* W1  = (const float*)d_in[1];
    const float* b1  = (const float*)d_in[2];
    const float* g1  = (const float*)d_in[3];
    const float* be1 = (const float*)d_in[4];
    const float* rm1 = (const float*)d_in[5];
    const float* rv1 = (const float*)d_in[6];
    const float* W2  = (const float*)d_in[7];
    const float* b2  = (const float*)d_in[8];
    const float* g2  = (const float*)d_in[9];
    const float* be2 = (const float*)d_in[10];
    const float* rm2 = (const float*)d_in[11];
    const float* rv2 = (const float*)d_in[12];
    const float* W3  = (const float*)d_in[13];
    const float* b3  = (const float*)d_in[14];
    const float* cb  = (const float*)d_in[15];
    const float* W4  = (const float*)d_in[16];
    const float* b4  = (const float*)d_in[17];
    const float* g3  = (const float*)d_in[18];
    const float* be3 = (const float*)d_in[19];
    const float* rm3 = (const float*)d_in[20];
    const float* rv3 = (const float*)d_in[21];
    const float* W5  = (const float*)d_in[22];
    const float* b5  = (const float*)d_in[23];
    const float* g4  = (const float*)d_in[24];
    const float* be4 = (const float*)d_in[25];
    const float* rm4 = (const float*)d_in[26];
    const float* rv4 = (const float*)d_in[27];
    const float* W6  = (const float*)d_in[28];
    const float* b6  = (const float*)d_in[29];

    // output layout: x_hat [8192*256] | res_s [8*8192*128] | ce_s [8*8192*128]
    float* xhat  = (float*)d_out;
    float* res_s = xhat + (size_t)NROWS * INDIM;
    float* ce_s  = res_s + (size_t)MBOOK * NROWS * CDIM;

    // workspace carve-up (~24 MB)
    float* h1 = (float*)d_ws;                      // [8192,128] (reused as di)
    float* h2 = h1 + (size_t)NROWS * CDIM;         // [8192,256]
    float* ze = h2 + (size_t)NROWS * INDIM;        // [8192,128]
    float* rA = ze + (size_t)NROWS * CDIM;         // [8192,128]
    float* rB = rA + (size_t)NROWS * CDIM;         // [8192,128]
    float* ee = rB + (size_t)NROWS * CDIM;         // [8192]
    int*   idx = (int*)(ee + KCODES);              // [8192]

    const int gemm128 = (NROWS / 16) * (CDIM / 16) / 4;   // 1024 blocks
    const int gemm256 = (NROWS / 16) * (INDIM / 16) / 4;  // 2048 blocks

    // encoder
    dense_wmma<<<gemm128, 128, 0, stream>>>(x,  W1, b1, g1, be1, rm1, rv1, h1, INDIM, CDIM, 1);
    dense_wmma<<<gemm256, 128, 0, stream>>>(h1, W2, b2, g2, be2, rm2, rv2, h2, CDIM, INDIM, 1);
    dense_wmma<<<gemm128, 128, 0, stream>>>(h2, W3, b3, nullptr, nullptr, nullptr, nullptr, ze, INDIM, CDIM, 0);

    // residual VQ (8 sequential stages)
    const float* cur = ze;
    for (int m = 0; m < MBOOK; ++m) {
        const float* Em = cb + (size_t)m * KCODES * CDIM;
        ee_norm  <<<KCODES / 256, 256, 0, stream>>>(Em, ee);
        vq_argmin<<<NROWS / 64, 128, 0, stream>>>(cur, Em, ee, idx);
        float* nxt = (m & 1) ? rB : rA;
        vq_finish<<<NROWS * (CDIM / 4) / 256, 256, 0, stream>>>(
            cur, Em, idx,
            res_s + (size_t)m * NROWS * CDIM,
            ce_s  + (size_t)m * NROWS * CDIM,
            nxt);
        cur = nxt;
    }

    // di = zq = ze - res_final   (straight-through forward value)
    di_sub<<<NROWS * CDIM / 4 / 256, 256, 0, stream>>>(ze, cur, h1);

    // decoder
    dense_wmma<<<gemm256, 128, 0, stream>>>(h1, W4, b4, g3, be3, rm3, rv3, h2, CDIM, INDIM, 1);
    dense_wmma<<<gemm128, 128, 0, stream>>>(h2, W5, b5, g4, be4, rm4, rv4, rA, INDIM, CDIM, 1);
    dense_wmma<<<gemm256, 128, 0, stream>>>(rA, W6, b6, nullptr, nullptr, nullptr, nullptr, xhat, CDIM, INDIM, 0);
}